// DpSe_9380208575075
// MI455X (gfx1250) — compile-verified
//
#include <hip/hip_runtime.h>
#include <hip/hip_bf16.h>

// ---------------------------------------------------------------------------
// DP-SE energy + force, fused per 16-atom tile. WMMA f32 16x16x4 for the fit
// network GEMMs (fwd and transposed bwd). Wave32 / gfx1250.
// GEMM inner loops are grouped (4 K-steps): fragment loads cluster ahead of
// the WMMA burst; the second-N-tile predicate is hoisted out of the K-loops.
// ---------------------------------------------------------------------------

#define B_   4
#define N_   2048
#define M_   128
#define NTY  2
#define GD   100
#define EH0  25
#define EH1  50
#define FIN  1600
#define FH   240
#define AT   16
#define NTHREADS 256
#define RMIN_ 0.5f
#define RMAX_ 6.0f
#define PI_   3.1415926535f
#define INV_MN (1.0f/(float)(M_*NTY))

typedef float v2f __attribute__((ext_vector_type(2)));
typedef float v8f __attribute__((ext_vector_type(8)));

__device__ __forceinline__ v8f wmma4(v8f c, v2f a, v2f b) {
  // D = A(16x4) * B(4x16) + C(16x16), all f32
  return __builtin_amdgcn_wmma_f32_16x16x4_f32(false, a, false, b, (short)0, c, false, false);
}
__device__ __forceinline__ v8f z8() {
  v8f z = {0.f,0.f,0.f,0.f,0.f,0.f,0.f,0.f};
  return z;
}
// gfx1250 native transcendental tanh (TRANS op, co-executes with VALU)
__device__ __forceinline__ float fast_tanh(float x) {
  float y;
  asm("v_tanh_f32 %0, %1" : "=v"(y) : "v"(x));
  return y;
}

struct SM {
  float xyz [AT][4][GD];   // scaled descriptor moments
  float dxyz[AT][4][GD];   // gradient wrt xyz
  float x0  [AT][FH];
  float x1  [AT][FH];
  float t2  [AT][FH];      // tanh(z2) (so x2 = x1 + t2)
  float g0b [AT][FH];      // grad scratch A
  float g1b [AT][FH];      // grad scratch B
  float Dblk[AT][16];      // K-block of D, regenerated per 16 flat-K
  float wtile[8][AT][16];  // per-wave dD output tile
  float evals[AT];
  int   ity[AT];
};

__device__ __forceinline__ void smoothfn(float r, float& s, float& dsr) {
  float safe = fmaxf(r, 1e-12f);
  float inv  = 1.0f / safe;
  float tt = (r - RMIN_) / (RMAX_ - RMIN_);
  float u  = 0.5f * __cosf(PI_ * tt) + 0.5f;
  if (r < RMIN_)      { s = inv;     dsr = -inv * inv; }
  else if (r < RMAX_) { s = u * inv; dsr = -u * inv * inv
                              - 0.5f * PI_ * __sinf(PI_ * tt) * inv / (RMAX_ - RMIN_); }
  else                { s = 0.f;     dsr = 0.f; }
  if (!(r > 1e-8f))   { s = 0.f;     dsr = 0.f; }
}

// GEMM: acc += A(16xFH, LDS) @ B, B row-major (k*FH + col). Grouped x4 steps.
template <bool HAS1>
__device__ __forceinline__ void gemm_rowB_t(const float* A, const float* Bm,
                                            int col0, int col1,
                                            int lm, int lh, v8f& acc0, v8f& acc1) {
  const int abase = lm * FH + 2 * lh;
  for (int sg = 0; sg < FH / 16; ++sg) {
    v2f a[4], b[4], b2[4];
    #pragma unroll
    for (int u = 0; u < 4; ++u) {
      int k0 = (sg * 4 + u) * 4;
      int ka = k0 + 2 * lh;
      a[u] = *(const v2f*)(A + abase + k0);                 // ds_load_b64
      b[u].x = Bm[ka * FH + col0]; b[u].y = Bm[(ka + 1) * FH + col0];
      if (HAS1) { b2[u].x = Bm[ka * FH + col1]; b2[u].y = Bm[(ka + 1) * FH + col1]; }
    }
    #pragma unroll
    for (int u = 0; u < 4; ++u) {
      acc0 = wmma4(acc0, a[u], b[u]);
      if (HAS1) acc1 = wmma4(acc1, a[u], b2[u]);
    }
  }
}
__device__ __forceinline__ void gemm_rowB(const float* A, const float* Bm,
                                          int col0, int col1, bool has1,
                                          int lm, int lh, v8f& acc0, v8f& acc1) {
  if (has1) gemm_rowB_t<true >(A, Bm, col0, col1, lm, lh, acc0, acc1);
  else      gemm_rowB_t<false>(A, Bm, col0, col1, lm, lh, acc0, acc1);
}

// GEMM: acc += A(16xFH, LDS) @ B^T, element (k,col) = Bm[col*FH + k].
// B-frag is one global_load_b64. Grouped x4 steps.
template <bool HAS1>
__device__ __forceinline__ void gemm_colB_t(const float* A, const float* Bm,
                                            int col0, int col1,
                                            int lm, int lh, v8f& acc0, v8f& acc1) {
  const int abase = lm * FH + 2 * lh;
  const float* bp0 = Bm + col0 * FH + 2 * lh;
  const float* bp1 = Bm + col1 * FH + 2 * lh;
  for (int sg = 0; sg < FH / 16; ++sg) {
    v2f a[4], b[4], b2[4];
    #pragma unroll
    for (int u = 0; u < 4; ++u) {
      int k0 = (sg * 4 + u) * 4;
      a[u] = *(const v2f*)(A + abase + k0);                 // ds_load_b64
      b[u] = *(const v2f*)(bp0 + k0);                       // global_load_b64
      if (HAS1) b2[u] = *(const v2f*)(bp1 + k0);
    }
    #pragma unroll
    for (int u = 0; u < 4; ++u) {
      acc0 = wmma4(acc0, a[u], b[u]);
      if (HAS1) acc1 = wmma4(acc1, a[u], b2[u]);
    }
  }
}
__device__ __forceinline__ void gemm_colB(const float* A, const float* Bm,
                                          int col0, int col1, bool has1,
                                          int lm, int lh, v8f& acc0, v8f& acc1) {
  if (has1) gemm_colB_t<true >(A, Bm, col0, col1, lm, lh, acc0, acc1);
  else      gemm_colB_t<false>(A, Bm, col0, col1, lm, lh, acc0, acc1);
}

__global__ void init_out_kernel(float* __restrict__ out) {
  int i = blockIdx.x * blockDim.x + threadIdx.x;
  int stride = gridDim.x * blockDim.x;
  for (int j = i; j < 4; j += stride) out[j] = 0.f;                   // Etot
  const int fbase = 4 + B_ * N_;
  for (int j = i; j < B_ * N_ * 3; j += stride) out[fbase + j] = 0.f; // F
}

__global__ __launch_bounds__(NTHREADS, 1)
void dp_se_kernel(const int*   __restrict__ Imagetype,
                  const int*   __restrict__ nbl,
                  const int*   __restrict__ nbt,
                  const float* __restrict__ ImagedR,
                  const float* __restrict__ eW0, const float* __restrict__ eb0,
                  const float* __restrict__ eW1, const float* __restrict__ eb1,
                  const float* __restrict__ eW2, const float* __restrict__ eb2,
                  const float* __restrict__ fW0, const float* __restrict__ fb0,
                  const float* __restrict__ fW1, const float* __restrict__ fb1,
                  const float* __restrict__ fW2, const float* __restrict__ fb2,
                  const float* __restrict__ fW3, const float* __restrict__ fb3,
                  float* __restrict__ out) {
  __shared__ SM sm;
  const int t    = threadIdx.x;
  const int lane = t & 31;
  const int wave = __builtin_amdgcn_readfirstlane(t >> 5);  // scalar wave id
  const int lm   = lane & 15;
  const int lh   = lane >> 4;
  const int aBase = blockIdx.x * AT;
  const int batch = aBase / N_;

  if (t < AT) sm.ity[t] = Imagetype[aBase + t];
  for (int idx = t; idx < AT * 4 * GD; idx += NTHREADS) {
    ((float*)sm.xyz)[idx]  = 0.f;
    ((float*)sm.dxyz)[idx] = 0.f;
  }
  __syncthreads();

  // ------------------- Phase 1: embedding forward, xyz accumulation --------
  for (int pit = 0; pit < (AT * M_) / NTHREADS; ++pit) {
    int pid = pit * NTHREADS + t;
    int a = pid >> 7;            // /128
    int m = pid & (M_ - 1);
    int gatom = aBase + a;
    int base  = gatom * M_ + m;
    float4 rv = ((const float4*)ImagedR)[base];
    float r = rv.x, dxv = rv.y, dyv = rv.z, dzv = rv.w;
    int jt = nbt[base];
    int it = sm.ity[a];
    float s, dsr; smoothfn(r, s, dsr);
    float safe = fmaxf(r, 1e-12f);
    float feat[4] = {s, s * dxv / safe, s * dyv / safe, s * dzv / safe};
    const float* W0p = eW0 + (it * NTY + jt) * EH0;
    const float* b0p = eb0 + (it * NTY + jt) * EH0;
    const float* W1p = eW1 + (it * NTY + jt) * EH0 * EH1;
    const float* b1p = eb1 + (it * NTY + jt) * EH1;
    const float* W2p = eW2 + (it * NTY + jt) * EH1 * GD;
    const float* b2p = eb2 + (it * NTY + jt) * GD;
    float a25[EH0];
    for (int j = 0; j < EH0; ++j) a25[j] = fast_tanh(s * W0p[j] + b0p[j]);
    float xc[EH1];
    for (int k = 0; k < EH1; ++k) {
      float z = b1p[k];
      #pragma unroll
      for (int j = 0; j < EH0; ++j) z += a25[j] * W1p[j * EH1 + k];
      xc[k] = fast_tanh(z) + a25[k < EH0 ? k : k - EH0];
    }
    for (int g = 0; g < GD; ++g) {
      float z = b2p[g];
      #pragma unroll
      for (int k = 0; k < EH1; ++k) z += xc[k] * W2p[k * GD + g];
      float Gg = fast_tanh(z) + xc[g < EH1 ? g : g - EH1];
      #pragma unroll
      for (int c = 0; c < 4; ++c) atomicAdd(&sm.xyz[a][c][g], feat[c] * Gg);
    }
  }
  __syncthreads();
  for (int idx = t; idx < AT * 4 * GD; idx += NTHREADS) ((float*)sm.xyz)[idx] *= INV_MN;
  __syncthreads();

  // ------------------- Phase 2+3: fit network fwd + bwd per type (WMMA) ----
  const int tile0 = wave;
  const int tile1 = wave + 8;
  const bool has1 = (tile1 < FH / 16);    // scalar: 15 N-tiles of 16
  const int col0  = tile0 * 16 + lm;
  const int col1  = tile1 * 16 + lm;

  for (int ti = 0; ti < NTY; ++ti) {
    const float* W0 = fW0 + ti * FIN * FH;
    const float* b0 = fb0 + ti * FH;
    const float* W1 = fW1 + ti * FH * FH;
    const float* b1 = fb1 + ti * FH;
    const float* W2 = fW2 + ti * FH * FH;
    const float* b2 = fb2 + ti * FH;
    const float* W3 = fW3 + ti * FH;
    const float  b3 = fb3[ti];
    if (t < AT) sm.evals[t] = 0.f;

    // x0 = tanh(D @ W0 + b0), D regenerated in 16-wide K blocks
    v8f acc0 = z8(), acc1 = z8();
    for (int kb = 0; kb < GD; ++kb) {
      __syncthreads();
      { int a = t >> 4, kk = t & 15;
        float sum = 0.f;
        #pragma unroll
        for (int c = 0; c < 4; ++c) sum += sm.xyz[a][c][kb] * sm.xyz[a][c][kk];
        sm.Dblk[a][kk] = sum; }
      __syncthreads();
      if (has1) {
        v2f av[4], bv[4], bv2[4];
        #pragma unroll
        for (int s4 = 0; s4 < 4; ++s4) {
          int ka = 4 * s4 + 2 * lh;
          av[s4] = *(const v2f*)(&sm.Dblk[lm][ka]);         // ds_load_b64
          int fa = kb * 16 + ka;
          bv[s4].x  = W0[fa * FH + col0]; bv[s4].y  = W0[(fa + 1) * FH + col0];
          bv2[s4].x = W0[fa * FH + col1]; bv2[s4].y = W0[(fa + 1) * FH + col1];
        }
        #pragma unroll
        for (int s4 = 0; s4 < 4; ++s4) {
          acc0 = wmma4(acc0, av[s4], bv[s4]);
          acc1 = wmma4(acc1, av[s4], bv2[s4]);
        }
      } else {
        v2f av[4], bv[4];
        #pragma unroll
        for (int s4 = 0; s4 < 4; ++s4) {
          int ka = 4 * s4 + 2 * lh;
          av[s4] = *(const v2f*)(&sm.Dblk[lm][ka]);
          int fa = kb * 16 + ka;
          bv[s4].x = W0[fa * FH + col0]; bv[s4].y = W0[(fa + 1) * FH + col0];
        }
        #pragma unroll
        for (int s4 = 0; s4 < 4; ++s4) acc0 = wmma4(acc0, av[s4], bv[s4]);
      }
    }
    __syncthreads();
    #pragma unroll
    for (int rr = 0; rr < 8; ++rr) {
      int row = rr + 8 * lh;
      sm.x0[row][col0] = fast_tanh(acc0[rr] + b0[col0]);
      if (has1) sm.x0[row][col1] = fast_tanh(acc1[rr] + b0[col1]);
    }
    __syncthreads();

    // x1 = tanh(x0@W1 + b1) + x0
    acc0 = z8(); acc1 = z8();
    gemm_rowB(&sm.x0[0][0], W1, col0, col1, has1, lm, lh, acc0, acc1);
    #pragma unroll
    for (int rr = 0; rr < 8; ++rr) {
      int row = rr + 8 * lh;
      sm.x1[row][col0] = fast_tanh(acc0[rr] + b1[col0]) + sm.x0[row][col0];
      if (has1) sm.x1[row][col1] = fast_tanh(acc1[rr] + b1[col1]) + sm.x0[row][col1];
    }
    __syncthreads();

    // t2 = tanh(x1@W2 + b2)   (x2 = x1 + t2)
    acc0 = z8(); acc1 = z8();
    gemm_rowB(&sm.x1[0][0], W2, col0, col1, has1, lm, lh, acc0, acc1);
    #pragma unroll
    for (int rr = 0; rr < 8; ++rr) {
      int row = rr + 8 * lh;
      sm.t2[row][col0] = fast_tanh(acc0[rr] + b2[col0]);
      if (has1) sm.t2[row][col1] = fast_tanh(acc1[rr] + b2[col1]);
    }
    __syncthreads();

    // e = (x1+t2)@W3 + b3 ; gz2 = W3*(1-t2^2) -> g0b
    { int a = t >> 4, s_ = t & 15;
      float p = 0.f;
      for (int j = 0; j < FH / 16; ++j) {
        int n = s_ + 16 * j;
        p += (sm.x1[a][n] + sm.t2[a][n]) * W3[n];
      }
      atomicAdd(&sm.evals[a], p); }
    for (int idx = t; idx < AT * FH; idx += NTHREADS) {
      int a = idx / FH, n = idx % FH;
      float tv = sm.t2[a][n];
      sm.g0b[a][n] = W3[n] * (1.f - tv * tv);
    }
    __syncthreads();
    if (t < AT && sm.ity[t] == ti) {
      float e = sm.evals[t] + b3;
      out[4 + aBase + t] = e;                 // Ei
      unsafeAtomicAdd(&out[batch], e);        // Etot
    }

    // g1 = gz2@W2^T + W3 ; gz1 = g1*(1-t1^2) -> g1b ; g0b <- g1
    acc0 = z8(); acc1 = z8();
    gemm_colB(&sm.g0b[0][0], W2, col0, col1, has1, lm, lh, acc0, acc1);
    __syncthreads();
    #pragma unroll
    for (int rr = 0; rr < 8; ++rr) {
      int row = rr + 8 * lh;
      { float g1v = acc0[rr] + W3[col0];
        float t1v = sm.x1[row][col0] - sm.x0[row][col0];
        sm.g1b[row][col0] = g1v * (1.f - t1v * t1v);
        sm.g0b[row][col0] = g1v; }
      if (has1) {
        float g1v = acc1[rr] + W3[col1];
        float t1v = sm.x1[row][col1] - sm.x0[row][col1];
        sm.g1b[row][col1] = g1v * (1.f - t1v * t1v);
        sm.g0b[row][col1] = g1v; }
    }
    __syncthreads();

    // g0 = gz1@W1^T + g1 ; gz0 = g0*(1-x0^2) -> g0b
    acc0 = z8(); acc1 = z8();
    gemm_colB(&sm.g1b[0][0], W1, col0, col1, has1, lm, lh, acc0, acc1);
    __syncthreads();
    #pragma unroll
    for (int rr = 0; rr < 8; ++rr) {
      int row = rr + 8 * lh;
      { float g0v = acc0[rr] + sm.g0b[row][col0];
        float xv  = sm.x0[row][col0];
        sm.g0b[row][col0] = g0v * (1.f - xv * xv); }
      if (has1) {
        float g0v = acc1[rr] + sm.g0b[row][col1];
        float xv  = sm.x0[row][col1];
        sm.g0b[row][col1] = g0v * (1.f - xv * xv); }
    }
    __syncthreads();

    // dD tiles = gz0 @ W0^T, contracted directly into dxyz (type-masked)
    for (int n = wave; n < GD; n += 8) {
      v8f acc = z8(), accd = z8();
      gemm_colB_t<false>(&sm.g0b[0][0], W0, n * 16 + lm, 0, lm, lh, acc, accd);
      #pragma unroll
      for (int rr = 0; rr < 8; ++rr) sm.wtile[wave][rr + 8 * lh][lm] = acc[rr];
      // term: dxyz[a][c][h] += dD[a][n][h] * xyz[a][c][n]
      for (int p = 0; p < 8; ++p) {
        int idx = lane + 32 * p;
        int a = idx >> 4, h = idx & 15;
        if (sm.ity[a] == ti) {
          float v = sm.wtile[wave][a][h];
          #pragma unroll
          for (int c = 0; c < 4; ++c)
            atomicAdd(&sm.dxyz[a][c][h], v * sm.xyz[a][c][n]);
        }
      }
      // term: dxyz[a][c][n] += sum_h dD[a][n][h] * xyz[a][c][h]
      for (int p = 0; p < 2; ++p) {
        int idx = lane + 32 * p;
        int a = idx >> 2, c = idx & 3;
        if (sm.ity[a] == ti) {
          float ssum = 0.f;
          #pragma unroll
          for (int h = 0; h < 16; ++h) ssum += sm.wtile[wave][a][h] * sm.xyz[a][c][h];
          atomicAdd(&sm.dxyz[a][c][n], ssum);
        }
      }
    }
    __syncthreads();
  } // type loop

  for (int idx = t; idx < AT * 4 * GD; idx += NTHREADS) ((float*)sm.dxyz)[idx] *= INV_MN;
  __syncthreads();

  // ------------------- Phase 4: embedding backward + forces ----------------
  float* Fout = out + 4 + B_ * N_;
  for (int pit = 0; pit < (AT * M_) / NTHREADS; ++pit) {
    int pid = pit * NTHREADS + t;
    int a = pid >> 7;
    int m = pid & (M_ - 1);
    int gatom = aBase + a;
    int base  = gatom * M_ + m;
    float4 rv = ((const float4*)ImagedR)[base];
    float r = rv.x;
    float d3[3] = {rv.y, rv.z, rv.w};
    int jt = nbt[base];
    int it = sm.ity[a];
    float s, dsr; smoothfn(r, s, dsr);
    float safe = fmaxf(r, 1e-12f);
    float feat[4] = {s, s * d3[0] / safe, s * d3[1] / safe, s * d3[2] / safe};
    const float* W0p = eW0 + (it * NTY + jt) * EH0;
    const float* b0p = eb0 + (it * NTY + jt) * EH0;
    const float* W1p = eW1 + (it * NTY + jt) * EH0 * EH1;
    const float* b1p = eb1 + (it * NTY + jt) * EH1;
    const float* W2p = eW2 + (it * NTY + jt) * EH1 * GD;
    const float* b2p = eb2 + (it * NTY + jt) * GD;
    float a25[EH0];
    for (int j = 0; j < EH0; ++j) a25[j] = fast_tanh(s * W0p[j] + b0p[j]);
    float xc[EH1];
    for (int k = 0; k < EH1; ++k) {
      float z = b1p[k];
      #pragma unroll
      for (int j = 0; j < EH0; ++j) z += a25[j] * W1p[j * EH1 + k];
      xc[k] = fast_tanh(z) + a25[k < EH0 ? k : k - EH0];
    }
    float dxc[EH1];
    for (int k = 0; k < EH1; ++k) dxc[k] = 0.f;
    float dfeat[4] = {0.f, 0.f, 0.f, 0.f};
    for (int g = 0; g < GD; ++g) {
      float z = b2p[g];
      #pragma unroll
      for (int k = 0; k < EH1; ++k) z += xc[k] * W2p[k * GD + g];
      float tg = fast_tanh(z);
      float Gg = tg + xc[g < EH1 ? g : g - EH1];
      float dGg = 0.f;
      #pragma unroll
      for (int c = 0; c < 4; ++c) {
        float dv = sm.dxyz[a][c][g];
        dGg += dv * feat[c];
        dfeat[c] += dv * Gg;
      }
      float dtg = dGg * (1.f - tg * tg);
      #pragma unroll
      for (int k = 0; k < EH1; ++k) dxc[k] += dtg * W2p[k * GD + g];
      dxc[g < EH1 ? g : g - EH1] += dGg;
    }
    float da25[EH0];
    for (int j = 0; j < EH0; ++j) da25[j] = 0.f;
    for (int k = 0; k < EH1; ++k) {
      float ht  = xc[k] - a25[k < EH0 ? k : k - EH0];
      float dz1 = dxc[k] * (1.f - ht * ht);
      #pragma unroll
      for (int j = 0; j < EH0; ++j) da25[j] += dz1 * W1p[j * EH1 + k];
    }
    for (int j = 0; j < EH0; ++j) da25[j] += dxc[j] + dxc[j + EH0];
    float dsE = 0.f;
    for (int j = 0; j < EH0; ++j) dsE += da25[j] * (1.f - a25[j] * a25[j]) * W0p[j];
    dfeat[0] += dsE;

    float inv1 = 1.f / safe, inv2 = inv1 * inv1, inv3 = inv2 * inv1;
    float fx[3];
    #pragma unroll
    for (int c = 0; c < 3; ++c) {
      float accf = dfeat[0] * (dsr * inv1) * d3[c];
      #pragma unroll
      for (int k = 0; k < 3; ++k) {
        float J = dsr * d3[k] * d3[c] * inv2
                + s * (((k == c) ? 1.f : 0.f) * inv1 - d3[k] * d3[c] * inv3);
        accf += dfeat[k + 1] * J;
      }
      fx[c] = accf;
    }
    #pragma unroll
    for (int c = 0; c < 3; ++c) unsafeAtomicAdd(&Fout[gatom * 3 + c], -fx[c]);
    int nl = nbl[base];
    if (nl > 0) {
      int rowN = batch * N_ + (nl - 1);
      #pragma unroll
      for (int c = 0; c < 3; ++c) unsafeAtomicAdd(&Fout[rowN * 3 + c], fx[c]);
    }
  }
}

extern "C" void kernel_launch(void* const* d_in, const int* in_sizes, int n_in,
                              void* d_out, int out_size, void* d_ws, size_t ws_size,
                              hipStream_t stream) {
  const int*   Imagetype = (const int*)  d_in[0];
  const int*   nbl       = (const int*)  d_in[1];
  const int*   nbt       = (const int*)  d_in[2];
  const float* ImagedR   = (const float*)d_in[3];
  // d_in[4] = nghost (0)
  const float* eW0 = (const float*)d_in[5];
  const float* eb0 = (const float*)d_in[6];
  const float* eW1 = (const float*)d_in[7];
  const float* eb1 = (const float*)d_in[8];
  const float* eW2 = (const float*)d_in[9];
  const float* eb2 = (const float*)d_in[10];
  const float* fW0 = (const float*)d_in[11];
  const float* fb0 = (const float*)d_in[12];
  const float* fW1 = (const float*)d_in[13];
  const float* fb1 = (const float*)d_in[14];
  const float* fW2 = (const float*)d_in[15];
  const float* fb2 = (const float*)d_in[16];
  const float* fW3 = (const float*)d_in[17];
  const float* fb3 = (const float*)d_in[18];
  float* out = (float*)d_out;

  init_out_kernel<<<64, 256, 0, stream>>>(out);
  dp_se_kernel<<<(B_ * N_) / AT, NTHREADS, 0, stream>>>(
      Imagetype, nbl, nbt, ImagedR,
      eW0, eb0, eW1, eb1, eW2, eb2,
      fW0, fb0, fW1, fb1, fW2, fb2, fW3, fb3, out);
}